// CircuitModel_54674933678366
// MI455X (gfx1250) — compile-verified
//
#include <hip/hip_runtime.h>

// CDNA5 / gfx1250: wave32, WMMA f32 16x16x4.
typedef __attribute__((ext_vector_type(2))) float v2f;
typedef __attribute__((ext_vector_type(8))) float v8f;

constexpr int B    = 64;
constexpr int T    = 256;
constexpr int NI   = 512;
constexpr int NO   = 512;
constexpr int NOBS = 128;
constexpr int CH   = 16;        // time-chunk size (matches WMMA N)
constexpr int NCH  = T / CH;    // 16 chunks

// LDS bank-conflict-free padded strides (64 banks x 4B):
constexpr int SW_S = 516;       // sW/sX row stride: 516 mod 64 = 4 -> 32 distinct bank pairs
constexpr int SX_S = 516;
constexpr int SP_S = 18;        // 18 mod 64: distinct banks for 32-lane scan reads + tile stores
constexpr int SY_S = 20;        // 20*lo mod 64 spaced by 4 -> conflict-free v2f A loads
constexpr int SS_S = 18;

#define ETA_LR 0.01f            // ETA * LR

__launch_bounds__(256, 1)
__global__ void hebb_scan_wmma(const float* __restrict__ X,
                               const float* __restrict__ Winit,
                               const int*   __restrict__ obs,
                               float*       __restrict__ out)
{
    // ~318 KB LDS: resident observed-row weights + chunk buffers (padded).
    __shared__ __align__(16) float sW[NOBS * SW_S];   // 258 KB
    __shared__ __align__(16) float sX[CH * SX_S];     //  32.25 KB  x chunk, [t][i]
    __shared__ __align__(16) float sP[NOBS * SP_S];   //   9 KB     base pre, [o][t]
    __shared__ __align__(16) float sY[NOBS * SY_S];   //  10 KB     ETA*y,   [o][t]
    __shared__ __align__(16) float sS[CH * SS_S];     //   1.1 KB   Gram,    [s][t]
    __shared__ int sObs[NOBS];

    const int b    = blockIdx.x;
    const int tid  = threadIdx.x;
    const int lane = tid & 31;
    const int wave = tid >> 5;      // 0..7
    const int lo   = lane & 15;     // 0..15
    const int hi   = lane >> 4;     // 0 or 1

    if (tid < NOBS) sObs[tid] = obs[tid];
    __syncthreads();

    // Gather observed rows of W_init[b] into LDS.
    const float* Wb = Winit + (size_t)b * NO * NI;
    for (int idx = tid; idx < NOBS * NI; idx += 256) {
        int o = idx >> 9;           // / NI
        int i = idx & (NI - 1);
        sW[o * SW_S + i] = Wb[(size_t)sObs[o] * NI + i];
    }
    __syncthreads();

    const float* Xb   = X   + (size_t)b * T * NI;
    float*       outB = out + (size_t)b * T * NOBS;

    for (int c = 0; c < NCH; ++c) {
        // ---- stage x chunk (16 x 512) ----
        for (int idx = tid; idx < CH * NI; idx += 256) {
            int t = idx >> 9;
            int i = idx & (NI - 1);
            sX[t * SX_S + i] = Xb[(size_t)c * CH * NI + idx];
        }
        // L2 prefetch of next chunk's X (32 KB, one cacheline per thread),
        // overlapped with this chunk's WMMA work.
        if (c + 1 < NCH)
            __builtin_prefetch(Xb + (size_t)(c + 1) * CH * NI + tid * 32, 0, 1);
        __syncthreads();

        // ---- P = W (128x512) @ Xc^T (512x16): wave w owns o-tile w ----
        {
            const int arow = (wave * 16 + lo) * SW_S;   // A row: M = lo
            const int brow = lo * SX_S;                 // B col: N = lo (t)
            v8f acc = {};
            for (int kb = 0; kb < NI; kb += 4) {
                int k0 = kb + 2 * hi;
                v2f a  = *(const v2f*)&sW[arow + k0];
                v2f bb = *(const v2f*)&sX[brow + k0];   // B[k][t] = Xc[t][k]
                acc = __builtin_amdgcn_wmma_f32_16x16x4_f32(
                          false, a, false, bb, (short)0, acc, false, false);
            }
            #pragma unroll
            for (int v = 0; v < 8; ++v)
                sP[(wave * 16 + v + 8 * hi) * SP_S + lo] = acc[v];
        }

        // ---- S = Xc @ Xc^T (16x16 Gram): wave 0 only; A == B operand ----
        if (wave == 0) {
            const int row = lo * SX_S;
            v8f s = {};
            for (int kb = 0; kb < NI; kb += 4) {
                int k0 = kb + 2 * hi;
                v2f a = *(const v2f*)&sX[row + k0];
                s = __builtin_amdgcn_wmma_f32_16x16x4_f32(
                        false, a, false, a, (short)0, s, false, false);
            }
            #pragma unroll
            for (int v = 0; v < 8; ++v)
                sS[(v + 8 * hi) * SS_S + lo] = s[v];
        }
        __syncthreads();

        // ---- in-chunk sequential scan: one thread per observed unit ----
        if (tid < NOBS) {
            const int o = tid;
            float ys[CH];
            float* dst = outB + (size_t)c * CH * NOBS;
            #pragma unroll
            for (int t = 0; t < CH; ++t) {
                float pre = sP[o * SP_S + t];
                for (int s = 0; s < t; ++s)
                    pre += ys[s] * sS[s * SS_S + t];   // ETA folded into ys
                float y = 1.0f / (1.0f + __expf(-pre));
                dst[t * NOBS + o] = y;                 // m[b, c*16+t, o]
                ys[t] = ETA_LR * y;
            }
            #pragma unroll
            for (int t = 0; t < CH; ++t)
                sY[o * SY_S + t] = ys[t];
        }
        __syncthreads();

        // ---- W += (ETA*Y)(128x16) @ Xc(16x512): wave w owns o-tile w ----
        {
            const int yrow = (wave * 16 + lo) * SY_S;
            for (int it = 0; it < NI / 16; ++it) {
                const int i0 = it * 16 + lo;           // B col: N = i
                v8f cacc;
                #pragma unroll
                for (int v = 0; v < 8; ++v)
                    cacc[v] = sW[(wave * 16 + v + 8 * hi) * SW_S + i0];
                #pragma unroll
                for (int kb = 0; kb < CH; kb += 4) {
                    int k0 = kb + 2 * hi;
                    v2f a = *(const v2f*)&sY[yrow + k0];
                    v2f bb;
                    bb.x = sX[k0 * SX_S + i0];         // B[s][i] = Xc[s][i]
                    bb.y = sX[(k0 + 1) * SX_S + i0];
                    cacc = __builtin_amdgcn_wmma_f32_16x16x4_f32(
                               false, a, false, bb, (short)0, cacc, false, false);
                }
                #pragma unroll
                for (int v = 0; v < 8; ++v)
                    sW[(wave * 16 + v + 8 * hi) * SW_S + i0] = cacc[v];
            }
        }
        __syncthreads();
    }
}

extern "C" void kernel_launch(void* const* d_in, const int* in_sizes, int n_in,
                              void* d_out, int out_size, void* d_ws, size_t ws_size,
                              hipStream_t stream) {
    const float* X    = (const float*)d_in[0];   // (B, T, NI) f32
    const float* W    = (const float*)d_in[1];   // (B, NO, NI) f32
    const int*   obsI = (const int*)d_in[2];     // (NOBS,) i32
    float*       out  = (float*)d_out;           // (B, T, NOBS) f32
    (void)in_sizes; (void)n_in; (void)out_size; (void)d_ws; (void)ws_size;

    hebb_scan_wmma<<<dim3(B), dim3(256), 0, stream>>>(X, W, obsI, out);
}